// BackprojectorModel_10411000725551
// MI455X (gfx1250) — compile-verified
//
#include <hip/hip_runtime.h>

// Voxel-driven backprojection. Gather-bound, L2-resident (28 MB total).
// CDNA5 path: per-lane async global->LDS staging (ASYNCcnt), double-buffered.
// LDS layout is slot-major [slot][batch] so one ds_load_b128 fetches all 4
// batches for a detector tap; interior tiles take a mask-free fast path.

#ifndef __has_builtin
#define __has_builtin(x) 0
#endif

#if __has_builtin(__builtin_amdgcn_global_load_async_to_lds_b32)
#define HAVE_ASYNC_LDS 1
#else
#define HAVE_ASYNC_LDS 0
#endif

#define BP_B 4
#define BP_NANG 720
#define BP_NDET 1024
#define BP_N 1024
#define TILE 32
#define SEG 64            // >= 31*(|c|+|s|)+2 = 46; 64 gives slack
#define THREADS 256

typedef __attribute__((address_space(1))) int gas_int;  // global
typedef __attribute__((address_space(3))) int las_int;  // LDS

// Segment start (block-uniform), clamped so [lo_c, lo_c+SEG-1] is in-bounds.
// All clamped tap indices provably land inside the staged window (span<=47<64).
__device__ __forceinline__ int seg_lo(float c, float s, int jx0, int iy0) {
  float t00  = ((float)jx0 - 511.5f) * c + ((float)iy0 - 511.5f) * s + 511.5f;
  float tmin = t00 + fminf(31.0f * c, 0.0f) + fminf(31.0f * s, 0.0f);
  int lo = (int)floorf(tmin);
  lo = lo < 0 ? 0 : lo;
  const int lomax = BP_NDET - SEG;
  return lo > lomax ? lomax : lo;
}

template <int Nle>
__device__ __forceinline__ void wait_async_le() {
#if HAVE_ASYNC_LDS
#if __has_builtin(__builtin_amdgcn_s_wait_asynccnt)
  __builtin_amdgcn_s_wait_asynccnt(Nle);
#else
  asm volatile("s_wait_asynccnt %0" ::"n"(Nle) : "memory");
#endif
#endif
}

// 256 threads stage 4 batches x 64 bins, one b32 each, TRANSPOSING into
// slot-major LDS [o][b] via per-lane async LDS addresses.
__device__ __forceinline__ void stage_seg(const float* __restrict__ y, int ang,
                                          int lo_c, float* dstBase, int tid) {
  const int b = tid >> 6;   // 0..3
  const int o = tid & 63;   // 0..63
  const float* gp = y + ((size_t)b * BP_NANG + (size_t)ang) * BP_NDET + (lo_c + o);
  float* lp = dstBase + o * BP_B + b;   // slot-major
#if HAVE_ASYNC_LDS
  __builtin_amdgcn_global_load_async_to_lds_b32(
      (gas_int*)(void*)gp,
      (las_int*)(void*)lp,
      /*offset=*/0, /*cpol=*/0);
#else
  *lp = *gp;
#endif
}

__global__ void __launch_bounds__(THREADS)
bp_kernel(const float* __restrict__ y, const float* __restrict__ angles,
          float* __restrict__ out) {
  __shared__ float s_cos[BP_NANG];
  __shared__ float s_sin[BP_NANG];
  __shared__ __align__(16) float s_seg[2][SEG][BP_B];  // slot-major, 2 buffers

  const int tid = threadIdx.x;

  // Per-angle cos/sin table (5.8 KB LDS), built once per block.
  for (int a = tid; a < BP_NANG; a += THREADS) {
    float th = angles[a];
    s_cos[a] = cosf(th);
    s_sin[a] = sinf(th);
  }
  __syncthreads();

  const int jx0 = blockIdx.x * TILE;
  const int iy0 = blockIdx.y * TILE;
  const int lx  = tid & (TILE - 1);   // lane -> consecutive j (coalesced)
  const int ly  = tid >> 5;           // wave row; strips at ly + 8k
  const int j   = jx0 + lx;
  const float xw  = (float)j - 511.5f;
  const float yw0 = (float)(iy0 + ly) - 511.5f;

  float acc[BP_B][4];
#pragma unroll
  for (int b = 0; b < BP_B; ++b)
#pragma unroll
    for (int k = 0; k < 4; ++k) acc[b][k] = 0.0f;

  // Prime buffer 0 with angle 0.
  stage_seg(y, 0, seg_lo(s_cos[0], s_sin[0], jx0, iy0), &s_seg[0][0][0], tid);

  for (int a = 0; a < BP_NANG; ++a) {
    const int buf = a & 1;

    // Issue next angle's staging into the other buffer; async loads complete
    // in order, so asynccnt<=1 guarantees angle-a data is resident.
    if (a + 1 < BP_NANG) {
      stage_seg(y, a + 1, seg_lo(s_cos[a + 1], s_sin[a + 1], jx0, iy0),
                &s_seg[buf ^ 1][0][0], tid);
      wait_async_le<1>();
    } else {
      wait_async_le<0>();
    }
    __syncthreads();

    const float c = s_cos[a];
    const float s = s_sin[a];

    // Block-uniform window bounds for this angle.
    const float t00  = ((float)jx0 - 511.5f) * c + ((float)iy0 - 511.5f) * s + 511.5f;
    const float dc   = 31.0f * c;
    const float dsn  = 31.0f * s;
    const float tmin = t00 + fminf(dc, 0.0f) + fminf(dsn, 0.0f);
    const float tmax = t00 + fmaxf(dc, 0.0f) + fmaxf(dsn, 0.0f);
    int lo = (int)floorf(tmin);
    lo = lo < 0 ? 0 : lo;
    const int lo_c = lo > (BP_NDET - SEG) ? (BP_NDET - SEG) : lo;
    const bool interior = (tmin >= 0.0f) && (tmax < 1023.0f);

    const float4* segv = (const float4*)&s_seg[buf][0][0];

    float t = xw * c + yw0 * s + 511.5f;   // detector coord for strip k=0
    const float tstep = 8.0f * s;          // y-strip stride = 8 rows

    if (interior) {
      // No edge masks/clamps possible: i0>=0 and i0+1<=1023 for every voxel.
#pragma unroll
      for (int k = 0; k < 4; ++k) {
        const float fi = floorf(t);
        const int   i0 = (int)fi;
        const float w  = t - fi;
        const int s0i  = i0 - lo_c;        // in [0, 62]
        const float4 v0 = segv[s0i];
        const float4 v1 = segv[s0i + 1];   // ds_load_b128 offset:16
        const float w0 = 1.0f - w;
        acc[0][k] = fmaf(w0, v0.x, fmaf(w, v1.x, acc[0][k]));
        acc[1][k] = fmaf(w0, v0.y, fmaf(w, v1.y, acc[1][k]));
        acc[2][k] = fmaf(w0, v0.z, fmaf(w, v1.z, acc[2][k]));
        acc[3][k] = fmaf(w0, v0.w, fmaf(w, v1.w, acc[3][k]));
        t += tstep;
      }
    } else {
      // Edge tiles: reference-equivalent masks; clamped slots stay in window.
#pragma unroll
      for (int k = 0; k < 4; ++k) {
        const float fi = floorf(t);
        const int   i0 = (int)fi;
        const float w  = t - fi;
        const float m0 = (i0 >= 0 && i0 < BP_NDET) ? 1.0f : 0.0f;
        const float m1 = (i0 >= -1 && i0 < BP_NDET - 1) ? 1.0f : 0.0f;
        const int s0i = min(max(i0, 0), BP_NDET - 1) - lo_c;
        const int s1i = min(max(i0 + 1, 0), BP_NDET - 1) - lo_c;
        const float4 v0 = segv[s0i];
        const float4 v1 = segv[s1i];
        const float w0 = (1.0f - w) * m0;
        const float w1 = w * m1;
        acc[0][k] = fmaf(w0, v0.x, fmaf(w1, v1.x, acc[0][k]));
        acc[1][k] = fmaf(w0, v0.y, fmaf(w1, v1.y, acc[1][k]));
        acc[2][k] = fmaf(w0, v0.z, fmaf(w1, v1.z, acc[2][k]));
        acc[3][k] = fmaf(w0, v0.w, fmaf(w1, v1.w, acc[3][k]));
        t += tstep;
      }
    }
    __syncthreads();   // before next iteration overwrites this buffer
  }

  const float wscale = 0.0043633231299858240f;  // pi / 720
  const int iy = iy0 + ly;
#pragma unroll
  for (int b = 0; b < BP_B; ++b)
#pragma unroll
    for (int k = 0; k < 4; ++k)
      out[((size_t)b * BP_N + (size_t)(iy + 8 * k)) * BP_N + j] =
          acc[b][k] * wscale;
}

extern "C" void kernel_launch(void* const* d_in, const int* in_sizes, int n_in,
                              void* d_out, int out_size, void* d_ws, size_t ws_size,
                              hipStream_t stream) {
  (void)in_sizes; (void)n_in; (void)d_ws; (void)ws_size; (void)out_size;
  const float* y      = (const float*)d_in[0];   // [4, 720, 1024] f32
  const float* angles = (const float*)d_in[1];   // [720] f32
  float* out = (float*)d_out;                    // [4, 1024, 1024] f32

  dim3 grid(BP_N / TILE, BP_N / TILE);           // 32 x 32 = 1024 blocks
  bp_kernel<<<grid, THREADS, 0, stream>>>(y, angles, out);
}